// MHA_43198781063277
// MI455X (gfx1250) — compile-verified
//
#include <hip/hip_runtime.h>

#define HIDDEN   1024
#define HEADS    16
#define HEAD_DIM 64
#define BATCH    4
#define SEQ      2048
#define MTOK     (BATCH * SEQ)   // 8192 tokens

#define KC   32      // k-chunk staged per LDS buffer
#define WROW 136     // padded LDS row stride in floats (2*WROW % 64 == 16 -> no bank conflicts)

typedef float v2f __attribute__((ext_vector_type(2)));
typedef float v8f __attribute__((ext_vector_type(8)));

__device__ __forceinline__ v2f ld_g2(const float* p) { return *(const v2f*)p; }

// D = A(16x4) * B(4x16) + C, full fp32 on the matrix pipe.
__device__ __forceinline__ v8f wmma_f32(v2f a, v2f b, v8f c) {
    return __builtin_amdgcn_wmma_f32_16x16x4_f32(
        /*neg_a=*/false, a, /*neg_b=*/false, b,
        /*c_mod=*/(short)0, c, /*reuse_a=*/false, /*reuse_b=*/false);
}

// Async copy 16B global -> LDS (ASYNCcnt-tracked, no VGPR data round trip).
__device__ __forceinline__ void async_cp16(unsigned lds_byte_off, const float* gsrc) {
    asm volatile("global_load_async_to_lds_b128 %0, %1, off"
                 :: "v"(lds_byte_off), "v"(gsrc) : "memory");
}
__device__ __forceinline__ void wait_async0() {
    asm volatile("s_wait_asynccnt 0" ::: "memory");
}

// ---------------------------------------------------------------------------
// C[M,1024] = A[M,1024] @ W[1024,1024] + bias   (W stored [in,out], row-major)
// Block: 256 threads (8 waves), tile 128m x 128n. Wave tile 32m x 64n.
// W panels double-buffered in LDS via async global->LDS copies.
// ---------------------------------------------------------------------------
__global__ __launch_bounds__(256)
void gemm1024_bias(const float* __restrict__ A, const float* __restrict__ W,
                   const float* __restrict__ bias, float* __restrict__ C) {
    __shared__ float Wb[2][KC][WROW];

    const int tid  = threadIdx.x;
    const int lane = tid & 31;
    const int wave = tid >> 5;             // 0..7
    const int wm   = wave >> 1;            // 0..3
    const int wn   = wave & 1;             // 0..1
    const int m0   = blockIdx.x * 128 + wm * 32;
    const int nblk = blockIdx.y * 128;
    const int n0w  = wn * 64;
    const int lm   = lane & 15;            // A: M row / B,C: N col
    const int half = lane >> 4;
    const int kb   = half * 2;             // K sub-offset per ISA fragment layout

    const float* Arow0 = A + (size_t)(m0 + lm) * HIDDEN + kb;
    const float* Arow1 = Arow0 + (size_t)16 * HIDDEN;

    // Stage one 32k x 128n panel of W into Wb[buf]: 1024 x 16B chunks, 4/thread.
    auto stage = [&](int c, int buf) {
        const int k0c = c * KC;
#pragma unroll
        for (int j = 0; j < 4; ++j) {
            const int ch = tid + 256 * j;        // 0..1023
            const int kk = ch >> 5;              // 0..31
            const int nn = (ch & 31) * 4;        // 0..124
            const float* g = W + (size_t)(k0c + kk) * HIDDEN + nblk + nn;
            const unsigned lds = (unsigned)(unsigned long long)&Wb[buf][kk][nn];
            async_cp16(lds, g);
        }
    };

    v8f acc[2][4] = {};

    stage(0, 0);
    wait_async0();
    __syncthreads();

    const int NCk = HIDDEN / KC;
    for (int c = 0; c < NCk; ++c) {
        const int buf = c & 1;
        if (c + 1 < NCk) stage(c + 1, buf ^ 1);   // overlaps with compute below
        const int k0c = c * KC;
#pragma unroll
        for (int ks = 0; ks < KC / 4; ++ks) {
            const int kk = ks * 4 + kb;
            v2f a0 = ld_g2(Arow0 + k0c + ks * 4);
            v2f a1 = ld_g2(Arow1 + k0c + ks * 4);
#pragma unroll
            for (int t = 0; t < 4; ++t) {
                const int n = n0w + t * 16 + lm;
                v2f b = { Wb[buf][kk][n], Wb[buf][kk + 1][n] };
                acc[0][t] = wmma_f32(a0, b, acc[0][t]);
                acc[1][t] = wmma_f32(a1, b, acc[1][t]);
            }
        }
        wait_async0();       // our async writes into buf^1 have landed in LDS
        __syncthreads();     // all waves done reading buf before next overwrite
    }

    float bn[4];
#pragma unroll
    for (int t = 0; t < 4; ++t) bn[t] = bias[nblk + n0w + t * 16 + lm];
    const int mrow = 8 * half;
#pragma unroll
    for (int r = 0; r < 2; ++r)
#pragma unroll
        for (int t = 0; t < 4; ++t)
#pragma unroll
            for (int i = 0; i < 8; ++i) {
                const int m = m0 + r * 16 + mrow + i;
                C[(size_t)m * HIDDEN + nblk + n0w + t * 16 + lm] = acc[r][t][i] + bn[t];
            }
}

// ---------------------------------------------------------------------------
// Causal flash attention, fp32 WMMA. One wave per (b, head, 16-query tile).
// Block = 4 waves = 4 heads (equal work per wave). O written in place over Q.
// ---------------------------------------------------------------------------
__global__ __launch_bounds__(128)
void attn_flash(const float* __restrict__ Q, const float* __restrict__ K,
                const float* __restrict__ V, float* __restrict__ O) {
    __shared__ float Pbuf[4][16 * 16 + 8];    // per-wave P transpose staging

    const int lane = threadIdx.x & 31;
    const int wave = threadIdx.x >> 5;        // 0..3
    const int qt   = blockIdx.x;              // 0..127
    const int h    = blockIdx.y * 4 + wave;   // head
    const int b    = blockIdx.z;
    const int q0   = qt * 16;
    const int nkt  = qt + 1;                  // causal: k-tiles 0..qt

    const int lm   = lane & 15;
    const int half = lane >> 4;
    const int kb   = half * 2;
    const size_t headoff = (size_t)h * HEAD_DIM;

    // Preload Q tile as 16 A-fragments (head_dim 64 = 16 K-steps of 4).
    const float* qptr = Q + (size_t)(b * SEQ + q0 + lm) * HIDDEN + headoff + kb;
    v2f aq[16];
#pragma unroll
    for (int s = 0; s < 16; ++s) aq[s] = ld_g2(qptr + 4 * s);

    v8f o0 = {}, o1 = {}, o2 = {}, o3 = {};  // O tile: 16q x 64d
    float mrow[8], lrow[8];
#pragma unroll
    for (int i = 0; i < 8; ++i) { mrow[i] = -1.0e30f; lrow[i] = 0.0f; }

    float* P = &Pbuf[wave][0];

    for (int kt = 0; kt < nkt; ++kt) {
        const int k0 = kt * 16;

        // S tile = Q(16x64) * K^T(64x16): B[kd][key] = K[key][kd] -> float2 loads
        const float* kptr = K + (size_t)(b * SEQ + k0 + lm) * HIDDEN + headoff + kb;
        v8f acc = {};
#pragma unroll
        for (int s = 0; s < 16; ++s)
            acc = wmma_f32(aq[s], ld_g2(kptr + 4 * s), acc);

        // Online softmax. C-layout: lane holds key col (lm), rows 8*half+i.
#pragma unroll
        for (int i = 0; i < 8; ++i) {
            const int row = half * 8 + i;          // query row in tile
            const int kg  = k0 + lm;               // this lane's key
            float sv = (kg <= q0 + row) ? acc[i] * 0.125f : -1.0e30f;
            float tm = sv;                          // row max across 16-lane half
            tm = fmaxf(tm, __shfl_xor(tm, 1, 32));
            tm = fmaxf(tm, __shfl_xor(tm, 2, 32));
            tm = fmaxf(tm, __shfl_xor(tm, 4, 32));
            tm = fmaxf(tm, __shfl_xor(tm, 8, 32));
            const float mnew = fmaxf(mrow[i], tm);
            const float f    = __expf(mrow[i] - mnew);
            const float p    = __expf(sv - mnew);
            float ts = p;                           // row sum across half
            ts += __shfl_xor(ts, 1, 32);
            ts += __shfl_xor(ts, 2, 32);
            ts += __shfl_xor(ts, 4, 32);
            ts += __shfl_xor(ts, 8, 32);
            lrow[i] = lrow[i] * f + ts;
            mrow[i] = mnew;
            o0[i] *= f; o1[i] *= f; o2[i] *= f; o3[i] *= f;
            P[row * 16 + lm] = p;                   // C-layout -> [query][key]
        }

        // O += P(16x16) * V(16x64); P re-read in A layout (same-wave DS order)
#pragma unroll
        for (int s = 0; s < 4; ++s) {
            const int kk = k0 + 4 * s + kb;
            v2f ap = *(const v2f*)&P[lm * 16 + 4 * s + kb];
            const float* vbase = V + (size_t)(b * SEQ + kk) * HIDDEN + headoff + lm;
            v2f bv0 = { vbase[0],  vbase[HIDDEN] };
            v2f bv1 = { vbase[16], vbase[16 + HIDDEN] };
            v2f bv2 = { vbase[32], vbase[32 + HIDDEN] };
            v2f bv3 = { vbase[48], vbase[48 + HIDDEN] };
            o0 = wmma_f32(ap, bv0, o0);
            o1 = wmma_f32(ap, bv1, o1);
            o2 = wmma_f32(ap, bv2, o2);
            o3 = wmma_f32(ap, bv3, o3);
        }
    }

    // Normalize and store (in place over the Q region this wave owns).
    float* optr = O + (size_t)(b * SEQ + q0) * HIDDEN + headoff + lm;
#pragma unroll
    for (int i = 0; i < 8; ++i) {
        const int m = half * 8 + i;
        const float inv = 1.0f / lrow[i];
        optr[(size_t)m * HIDDEN +  0] = o0[i] * inv;
        optr[(size_t)m * HIDDEN + 16] = o1[i] * inv;
        optr[(size_t)m * HIDDEN + 32] = o2[i] * inv;
        optr[(size_t)m * HIDDEN + 48] = o3[i] * inv;
    }
}

// ---------------------------------------------------------------------------
extern "C" void kernel_launch(void* const* d_in, const int* in_sizes, int n_in,
                              void* d_out, int out_size, void* d_ws, size_t ws_size,
                              hipStream_t stream) {
    (void)in_sizes; (void)n_in; (void)out_size; (void)ws_size;
    const float* x  = (const float*)d_in[0];
    const float* Wq = (const float*)d_in[1];
    const float* bq = (const float*)d_in[2];
    const float* Wk = (const float*)d_in[3];
    const float* bk = (const float*)d_in[4];
    const float* Wv = (const float*)d_in[5];
    const float* bv = (const float*)d_in[6];
    const float* Wo = (const float*)d_in[7];
    const float* bo = (const float*)d_in[8];
    float* out = (float*)d_out;

    float* Qb = (float*)d_ws;                       // 32 MB each
    float* Kb = Qb + (size_t)MTOK * HIDDEN;
    float* Vb = Kb + (size_t)MTOK * HIDDEN;

    dim3 gg(MTOK / 128, HIDDEN / 128, 1);
    gemm1024_bias<<<gg, 256, 0, stream>>>(x, Wq, bq, Qb);
    gemm1024_bias<<<gg, 256, 0, stream>>>(x, Wk, bk, Kb);
    gemm1024_bias<<<gg, 256, 0, stream>>>(x, Wv, bv, Vb);

    dim3 ga(SEQ / 16, HEADS / 4, BATCH);
    attn_flash<<<ga, 128, 0, stream>>>(Qb, Kb, Vb, /*O in place=*/Qb);

    gemm1024_bias<<<gg, 256, 0, stream>>>(Qb, Wo, bo, out);
}